// GeometricHyperplaneEnergy_43430709297915
// MI455X (gfx1250) — compile-verified
//
#include <hip/hip_runtime.h>
#include <hip/hip_bf16.h>

#define N_POINTS 65536
#define DIM 8
#define BATCH 256
#define PLANE_TOPK 128
#define ACTIVE_TOPK 64
#define NCHUNK 16   // point chunks for K3 (proj max/min)
#define ICHUNK 8    // point chunks for K5 (inactive term)
#define MARGIN_F 0.05f
#define CAP 2048
#define PREBIN 4076  // pre-collection threshold bin: p >= 4076/4096 ~= 0.99512+

typedef __attribute__((ext_vector_type(2))) float v2f;
typedef __attribute__((ext_vector_type(8))) float v8f;

// ---- workspace layout (floats) ----
// Region A: per batch row, stride 256 floats
#define ROWSTRIDE 256
#define OFF_VAL64 0    // top-64 weights
#define OFF_IDX64 64   // top-64 indices (int bits in float slots)
#define OFF_WSUM  128
#define OFF_ANORM 129
#define OFF_CARD  130
#define OFF_PLANE 131
#define OFF_FACET 132
#define OFF_MAXP  133
#define OFF_MINP  134
#define OFF_NRM   136  // 8 floats
#define OFF_COV   144  // 64 floats
// Region B: cross-block partials
#define REGB        (BATCH * ROWSTRIDE)
#define OFF_MAXPART (REGB)
#define OFF_MINPART (REGB + NCHUNK * BATCH)
#define OFF_IPARTP  (REGB + 2 * NCHUNK * BATCH)
#define OFF_IPARTN  (OFF_IPARTP + ICHUNK * BATCH)

// ------------------------------------------------------------------
// K1: fused single-pass top-k (histogram + optimistic pre-collection,
//     rare exact fallback pass) + bitonic sort + moments -> covariance.
// ------------------------------------------------------------------
__global__ __launch_bounds__(256) void k_topk(const float* __restrict__ prob,
                                              const float* __restrict__ points,
                                              float* __restrict__ ws) {
  __shared__ unsigned hist[4096];
  __shared__ float cval[CAP];
  __shared__ int cidx[CAP];
  __shared__ unsigned cnt;
  __shared__ int tbin_s;
  __shared__ int need2_s;
  __shared__ float red[256];
  const int tid = threadIdx.x;
  const int b = blockIdx.x;
  for (int i = tid; i < 4096; i += 256) hist[i] = 0u;
  if (tid == 0) cnt = 0u;
  __syncthreads();
  const float* row = prob + (size_t)b * N_POINTS;
  float lsum = 0.f;
  for (int i = tid; i < N_POINTS; i += 256) {
    float p = row[i];
    p = fminf(fmaxf(p, 1e-5f), 1.f - 1e-5f);
    lsum += p;
    int bn = (int)(p * 4096.f);
    bn = bn < 0 ? 0 : (bn > 4095 ? 4095 : bn);
    atomicAdd(&hist[bn], 1u);
    if (bn >= PREBIN) {  // optimistic collection during the same sweep
      unsigned k = atomicAdd(&cnt, 1u);
      if (k < (unsigned)CAP) { cval[k] = p; cidx[k] = i; }
    }
  }
  red[tid] = lsum;
  __syncthreads();
  for (int s = 128; s > 0; s >>= 1) {
    if (tid < s) red[tid] += red[tid + s];
    __syncthreads();
  }
  if (tid == 0) {
    unsigned acc = 0; int tb = 0;
    for (int bn = 4095; bn >= 0; --bn) {
      acc += hist[bn];
      if (acc >= PLANE_TOPK) { tb = bn; break; }
    }
    tbin_s = tb;
    // exact iff threshold bin edge >= pre-collection edge and no cap overflow
    need2_s = (tb < PREBIN) || (cnt > (unsigned)CAP) ? 1 : 0;
  }
  __syncthreads();
  const int tb = tbin_s;
  if (need2_s) {  // uniform branch; rare fallback: exact re-collection
    if (tid == 0) cnt = 0u;
    __syncthreads();
    for (int i = tid; i < N_POINTS; i += 256) {
      float p = row[i];
      p = fminf(fmaxf(p, 1e-5f), 1.f - 1e-5f);
      int bn = (int)(p * 4096.f);
      bn = bn < 0 ? 0 : (bn > 4095 ? 4095 : bn);
      if (bn >= tb) {
        unsigned k = atomicAdd(&cnt, 1u);
        if (k < (unsigned)CAP) { cval[k] = p; cidx[k] = i; }
      }
    }
    __syncthreads();
  }
  unsigned nc = cnt; if (nc > (unsigned)CAP) nc = CAP;
  for (int i = tid; i < CAP; i += 256)
    if ((unsigned)i >= nc) { cval[i] = -1e30f; cidx[i] = -1; }
  __syncthreads();
  // bitonic sort, descending by value (canonicalizes atomic order)
  for (int k = 2; k <= CAP; k <<= 1) {
    for (int j = k >> 1; j > 0; j >>= 1) {
      for (int t = tid; t < CAP; t += 256) {
        int ixj = t ^ j;
        if (ixj > t) {
          bool up = ((t & k) == 0);
          float a = cval[t], c2 = cval[ixj];
          bool sw = up ? (a < c2) : (a > c2);
          if (sw) {
            cval[t] = c2; cval[ixj] = a;
            int ti = cidx[t]; cidx[t] = cidx[ixj]; cidx[ixj] = ti;
          }
        }
      }
      __syncthreads();
    }
  }
  float* base = ws + (size_t)b * ROWSTRIDE;
  if (tid < ACTIVE_TOPK) {
    base[OFF_VAL64 + tid] = cval[tid];
    base[OFF_IDX64 + tid] = __int_as_float(cidx[tid]);
  }
  if (tid == 0) {
    float wsum = 0.f, mean[8], sm[8][8];
    for (int d = 0; d < 8; ++d) mean[d] = 0.f;
    for (int i = 0; i < 8; ++i)
      for (int j = 0; j < 8; ++j) sm[i][j] = 0.f;
    for (int e = 0; e < PLANE_TOPK; ++e) {
      int id = cidx[e];
      if (id < 0) continue;
      float w = cval[e];
      float pt[8];
      for (int d = 0; d < 8; ++d) pt[d] = points[(size_t)id * 8 + d];
      wsum += w;
      for (int d = 0; d < 8; ++d) mean[d] += w * pt[d];
      for (int i = 0; i < 8; ++i)
        for (int j = 0; j < 8; ++j) sm[i][j] += w * pt[i] * pt[j];
    }
    float wc = fmaxf(wsum, 1e-6f);
    for (int d = 0; d < 8; ++d) mean[d] /= wc;
    for (int i = 0; i < 8; ++i)
      for (int j = 0; j < 8; ++j)
        base[OFF_COV + i * 8 + j] = sm[i][j] / wc - mean[i] * mean[j];
    float an = 0.f;
    for (int e = 0; e < ACTIVE_TOPK; ++e) an += fmaxf(cval[e], 0.f);
    base[OFF_ANORM] = fmaxf(an, 1e-6f);
    base[OFF_WSUM] = wc;
    base[OFF_CARD] = red[0];
  }
}

// ------------------------------------------------------------------
// K2: 8x8 symmetric Jacobi eigensolver, one thread per batch row.
// ------------------------------------------------------------------
__global__ __launch_bounds__(256) void k_eigh(float* __restrict__ ws) {
  const int b = threadIdx.x;
  float* base = ws + (size_t)b * ROWSTRIDE;
  float A[8][8], V[8][8];
  for (int i = 0; i < 8; ++i)
    for (int j = 0; j < 8; ++j) {
      A[i][j] = base[OFF_COV + i * 8 + j];
      V[i][j] = (i == j) ? 1.f : 0.f;
    }
  for (int sweep = 0; sweep < 12; ++sweep) {
    for (int p = 0; p < 7; ++p)
      for (int q = p + 1; q < 8; ++q) {
        float apq = A[p][q];
        if (fabsf(apq) < 1e-12f) continue;
        float theta = (A[q][q] - A[p][p]) / (2.f * apq);
        float t = 1.f / (fabsf(theta) + sqrtf(theta * theta + 1.f));
        if (theta < 0.f) t = -t;
        float c = 1.f / sqrtf(t * t + 1.f);
        float s = t * c;
        for (int k = 0; k < 8; ++k) {  // A <- A*G
          float akp = A[k][p], akq = A[k][q];
          A[k][p] = c * akp - s * akq;
          A[k][q] = s * akp + c * akq;
        }
        for (int k = 0; k < 8; ++k) {  // A <- G^T*A
          float apk = A[p][k], aqk = A[q][k];
          A[p][k] = c * apk - s * aqk;
          A[q][k] = s * apk + c * aqk;
        }
        for (int k = 0; k < 8; ++k) {  // V <- V*G
          float vkp = V[k][p], vkq = V[k][q];
          V[k][p] = c * vkp - s * vkq;
          V[k][q] = s * vkp + c * vkq;
        }
      }
  }
  int i0 = 0; float l0 = A[0][0];
  for (int i = 1; i < 8; ++i)
    if (A[i][i] < l0) { l0 = A[i][i]; i0 = i; }
  float l1 = 3.4e38f;
  for (int i = 0; i < 8; ++i)
    if (i != i0 && A[i][i] < l1) l1 = A[i][i];
  base[OFF_PLANE] = l0;
  base[OFF_FACET] = l0 / (l1 + 1e-6f);
  for (int d = 0; d < 8; ++d) base[OFF_NRM + d] = V[d][i0];
}

// ------------------------------------------------------------------
// K3: proj = points @ normal^T via V_WMMA_F32_16X16X4_F32; point
//     segments staged to LDS with GLOBAL_LOAD_ASYNC_TO_LDS_B128
//     (ASYNCcnt path); per-batch running max/min -> chunk partials.
// ------------------------------------------------------------------
#define SEG_PTS 1024  // points per LDS staging segment (32 KB)
__global__ __launch_bounds__(256) void k_projmax(const float* __restrict__ points,
                                                 float* __restrict__ ws) {
  __shared__ float nrm[BATCH * 8];           // 8 KB
  __shared__ float ptile[SEG_PTS * 8];       // 32 KB staged point rows
  const int tid = threadIdx.x;
  for (int t = tid; t < BATCH * 8; t += 256) {
    int bb = t >> 3, d = t & 7;
    nrm[t] = ws[(size_t)bb * ROWSTRIDE + OFF_NRM + d];
  }
  __syncthreads();
  const int wave = tid >> 5;
  const int lane = tid & 31;
  const int lo = lane & 15;
  const int hi = lane >> 4;
  // B fragments (4x16 f32): b[0]=B[2*hi][n], b[1]=B[2*hi+1][n]; n=col
  v2f bf[2][2];
  for (int bt = 0; bt < 2; ++bt) {
    int col = (wave * 2 + bt) * 16 + lo;
    for (int kc = 0; kc < 2; ++kc) {
      bf[bt][kc].x = nrm[col * 8 + kc * 4 + 2 * hi];
      bf[bt][kc].y = nrm[col * 8 + kc * 4 + 2 * hi + 1];
    }
  }
  float vmax[2][8], vmin[2][8];
  for (int bt = 0; bt < 2; ++bt)
    for (int v = 0; v < 8; ++v) { vmax[bt][v] = -3.4e38f; vmin[bt][v] = 3.4e38f; }
  const int ptBase = blockIdx.x * (N_POINTS / NCHUNK);
  const float* gptr = points + (size_t)ptBase * 8;
  const unsigned ldsBase = (unsigned)(unsigned long long)&ptile[0];
  for (int seg = 0; seg < (N_POINTS / NCHUNK) / SEG_PTS; ++seg) {
    // async-stage SEG_PTS point rows (SEG_PTS*32 B) into LDS, 16B per op
    for (int it = 0; it < (SEG_PTS * 32 / 16) / 256; ++it) {
      int e = it * 256 + tid;
      unsigned laddr = ldsBase + (unsigned)(e * 16);
      unsigned goff = (unsigned)(seg * SEG_PTS * 32 + e * 16);
      asm volatile("global_load_async_to_lds_b128 %0, %1, %2"
                   :: "v"(laddr), "v"(goff), "s"(gptr)
                   : "memory");
    }
    asm volatile("s_wait_asynccnt 0" ::: "memory");
    __syncthreads();
    for (int pt = 0; pt < SEG_PTS / 16; ++pt) {
      // A fragment (16x4 f32): a[0]=A[lo][2*hi], a[1]=A[lo][2*hi+1]
      const float* pr = &ptile[(size_t)(pt * 16 + lo) * 8];
      v2f a0 = { pr[2 * hi], pr[2 * hi + 1] };          // K=0..3
      v2f a1 = { pr[4 + 2 * hi], pr[4 + 2 * hi + 1] };  // K=4..7
      for (int bt = 0; bt < 2; ++bt) {
        v8f c = {0.f, 0.f, 0.f, 0.f, 0.f, 0.f, 0.f, 0.f};
        c = __builtin_amdgcn_wmma_f32_16x16x4_f32(false, a0, false, bf[bt][0],
                                                  (short)0, c, false, false);
        c = __builtin_amdgcn_wmma_f32_16x16x4_f32(false, a1, false, bf[bt][1],
                                                  (short)0, c, false, false);
        for (int v = 0; v < 8; ++v) {
          vmax[bt][v] = fmaxf(vmax[bt][v], c[v]);
          vmin[bt][v] = fminf(vmin[bt][v], c[v]);
        }
      }
    }
    __syncthreads();  // all waves done with segment before restaging
  }
  float* maxpart = ws + OFF_MAXPART;
  float* minpart = ws + OFF_MINPART;
  for (int bt = 0; bt < 2; ++bt) {
    float mx = vmax[bt][0], mn = vmin[bt][0];
    for (int v = 1; v < 8; ++v) {
      mx = fmaxf(mx, vmax[bt][v]);
      mn = fminf(mn, vmin[bt][v]);
    }
    mx = fmaxf(mx, __shfl_xor(mx, 16, 32));
    mn = fminf(mn, __shfl_xor(mn, 16, 32));
    if (hi == 0) {
      int batch = (wave * 2 + bt) * 16 + lo;
      maxpart[blockIdx.x * BATCH + batch] = mx;
      minpart[blockIdx.x * BATCH + batch] = mn;
    }
  }
}

// ------------------------------------------------------------------
// K4: reduce chunk partials -> per-batch max/min of proj.
// ------------------------------------------------------------------
__global__ __launch_bounds__(256) void k_redmax(float* __restrict__ ws) {
  const int b = threadIdx.x;
  const float* maxpart = ws + OFF_MAXPART;
  const float* minpart = ws + OFF_MINPART;
  float mx = -3.4e38f, mn = 3.4e38f;
  for (int c = 0; c < NCHUNK; ++c) {
    mx = fmaxf(mx, maxpart[c * BATCH + b]);
    mn = fminf(mn, minpart[c * BATCH + b]);
  }
  float* base = ws + (size_t)b * ROWSTRIDE;
  base[OFF_MAXP] = mx;
  base[OFF_MINP] = mn;
}

// ------------------------------------------------------------------
// K5: inactive terms. 2-D grid (point-chunk x batch-group-of-8):
//     each points element amortized over 8 batches (L2 traffic /8),
//     deterministic per-chunk partials.
// ------------------------------------------------------------------
__global__ __launch_bounds__(256) void k_inactive(const float* __restrict__ prob,
                                                  const float* __restrict__ points,
                                                  float* __restrict__ ws) {
  __shared__ float red[256];
  const int tid = threadIdx.x;
  const int chunk = blockIdx.x;   // 0..ICHUNK-1
  const int bg = blockIdx.y;      // 0..BATCH/8-1
  float nv[8][8], mxp[8], mnp[8];
  for (int j = 0; j < 8; ++j) {
    const float* base = ws + (size_t)(bg * 8 + j) * ROWSTRIDE;
    for (int d = 0; d < 8; ++d) nv[j][d] = base[OFF_NRM + d];
    mxp[j] = base[OFF_MAXP];
    mnp[j] = base[OFF_MINP];
  }
  float ap[8], an[8];
  for (int j = 0; j < 8; ++j) { ap[j] = 0.f; an[j] = 0.f; }
  const int i0 = chunk * (N_POINTS / ICHUNK);
  for (int it = 0; it < (N_POINTS / ICHUNK) / 256; ++it) {
    const int i = i0 + it * 256 + tid;
    const float* pt = points + (size_t)i * 8;
    float pv[8];
    for (int d = 0; d < 8; ++d) pv[d] = pt[d];
    for (int j = 0; j < 8; ++j) {
      float p = prob[(size_t)(bg * 8 + j) * N_POINTS + i];
      p = fminf(fmaxf(p, 1e-5f), 1.f - 1e-5f);
      float w = 1.f - p;
      float pj = 0.f;
      for (int d = 0; d < 8; ++d) pj += pv[d] * nv[j][d];
      float sp = fmaxf(MARGIN_F + pj - mxp[j], 0.f);
      float sn = fmaxf(MARGIN_F - pj + mnp[j], 0.f);
      ap[j] += w * sp * sp;
      an[j] += w * sn * sn;
    }
  }
  float* ipartP = ws + OFF_IPARTP;
  float* ipartN = ws + OFF_IPARTN;
  for (int j = 0; j < 8; ++j) {
    red[tid] = ap[j];
    __syncthreads();
    for (int s = 128; s > 0; s >>= 1) {
      if (tid < s) red[tid] += red[tid + s];
      __syncthreads();
    }
    if (tid == 0) ipartP[chunk * BATCH + bg * 8 + j] = red[0];
    __syncthreads();
    red[tid] = an[j];
    __syncthreads();
    for (int s = 128; s > 0; s >>= 1) {
      if (tid < s) red[tid] += red[tid + s];
      __syncthreads();
    }
    if (tid == 0) ipartN[chunk * BATCH + bg * 8 + j] = red[0];
    __syncthreads();
  }
}

// ------------------------------------------------------------------
// K6: active terms over top-64 indices + final combine.
// ------------------------------------------------------------------
__global__ __launch_bounds__(64) void k_final(const float* __restrict__ points,
                                              const float* __restrict__ ws,
                                              float* __restrict__ out) {
  __shared__ float rp[64], rn[64];
  const int t = threadIdx.x;
  const int b = blockIdx.x;
  const float* base = ws + (size_t)b * ROWSTRIDE;
  const float maxp = base[OFF_MAXP];
  const float minp = base[OFF_MINP];
  float nv[8];
  for (int d = 0; d < 8; ++d) nv[d] = base[OFF_NRM + d];
  float w = base[OFF_VAL64 + t];
  int idx = __float_as_int(base[OFF_IDX64 + t]);
  float ap = 0.f, an = 0.f;
  if (idx >= 0) {
    const float* pt = points + (size_t)idx * 8;
    float pj = 0.f;
    for (int d = 0; d < 8; ++d) pj += pt[d] * nv[d];
    float dp = pj - maxp, dn = pj - minp;
    ap = w * dp * dp;
    an = w * dn * dn;
  }
  rp[t] = ap; rn[t] = an;
  __syncthreads();
  for (int s = 32; s > 0; s >>= 1) {
    if (t < s) { rp[t] += rp[t + s]; rn[t] += rn[t + s]; }
    __syncthreads();
  }
  if (t == 0) {
    float ipos = 0.f, ineg = 0.f;
    const float* ipartP = ws + OFF_IPARTP;
    const float* ipartN = ws + OFF_IPARTN;
    for (int c = 0; c < ICHUNK; ++c) {
      ipos += ipartP[c * BATCH + b];
      ineg += ipartN[c * BATCH + b];
    }
    float anorm = base[OFF_ANORM];
    float card = base[OFF_CARD];
    float inorm = fmaxf((float)N_POINTS - card, 1e-6f);
    // support term is identically zero (signed.max == 0 by construction)
    float bpos = (rp[0] / anorm) + 0.35f * (ipos / inorm);
    float bneg = (rn[0] / anorm) + 0.35f * (ineg / inorm);
    float bound = (bpos <= bneg) ? bpos : bneg;
    float deficit = fmaxf(26.0f - card, 0.f);
    out[b] = base[OFF_PLANE] + 8.0f * base[OFF_FACET] + 4.0f * bound +
             25.0f * deficit * deficit;
  }
}

extern "C" void kernel_launch(void* const* d_in, const int* in_sizes, int n_in,
                              void* d_out, int out_size, void* d_ws, size_t ws_size,
                              hipStream_t stream) {
  const float* prob = (const float*)d_in[0];    // (BATCH, N_POINTS) f32
  const float* points = (const float*)d_in[1];  // (N_POINTS, DIM) f32
  float* ws = (float*)d_ws;                     // needs ~304 KB
  float* out = (float*)d_out;                   // (BATCH,) f32

  k_topk<<<BATCH, 256, 0, stream>>>(prob, points, ws);
  k_eigh<<<1, 256, 0, stream>>>(ws);
  k_projmax<<<NCHUNK, 256, 0, stream>>>(points, ws);
  k_redmax<<<1, 256, 0, stream>>>(ws);
  k_inactive<<<dim3(ICHUNK, BATCH / 8), 256, 0, stream>>>(prob, points, ws);
  k_final<<<BATCH, 64, 0, stream>>>(points, ws, out);
}